// TimeStretch_42949672961457
// MI455X (gfx1250) — compile-verified
//
#include <hip/hip_runtime.h>
#include <math.h>

typedef __attribute__((ext_vector_type(2))) float v2f;
typedef __attribute__((ext_vector_type(8))) float v8f;

#define T_IN   1024
#define T_OUT  788          // ceil(1024 / 1.3)
#define NFREQ  1025
#define NBATCH 32
#define ROWS   (NBATCH * NFREQ)   // 32800
#define RATE_F 1.3f
#define WAVES_PER_BLOCK 8
#define LDS_PER_WAVE 272    // 256 phase slots + 16 column totals

__global__ __launch_bounds__(256) void phase_vocoder_kernel(
    const float* __restrict__ spec,
    const float* __restrict__ phase_advance,
    float* __restrict__ out)
{
    __shared__ float sm[WAVES_PER_BLOCK * LDS_PER_WAVE];

    const int lane = threadIdx.x & 31;
    const int wib  = threadIdx.x >> 5;
    const int row  = blockIdx.x * WAVES_PER_BLOCK + wib;
    if (row >= ROWS) return;               // uniform per wave (4100*8 == ROWS)

    float* pbuf = &sm[wib * LDS_PER_WAVE]; // [256] per-wave phase staging
    float* tbuf = pbuf + 256;              // [16]  per-wave column totals

    const int   f    = row % NFREQ;
    const float padv = phase_advance[f];
    const float2* __restrict__ spec2 = (const float2*)(spec + (size_t)row * (T_IN * 2));
    float2* __restrict__ out2        = (float2*)(out + (size_t)row * (T_OUT * 2));

    const float TWO_PI     = 6.28318530717958647692f;
    const float INV_TWO_PI = 0.15915494309189533577f;

    // phase_0 = angle of frame 0 (computed redundantly by all lanes)
    float2 s00 = spec2[0];
    float prev_delta = atan2f(s00.y, s00.x);   // cross-chunk shift carry
    float carry = 0.0f;                        // running cumsum carry

    const int  n         = lane & 15;          // column / M index (0..15)
    const bool hi        = lane >= 16;
    const int  koff      = hi ? 2 : 0;         // K interleave of 16x4 f32 fragments
    const int  mrow_base = hi ? 8 : 0;         // C/D: hi lanes hold rows M=8..15

    // Constant lower-triangular-ones A, split into 4 K-slices of 16x4.
    // A layout (32-bit 16x4): lanes 0-15 M=lane; v.x -> K = 4kk+koff, v.y -> K = 4kk+koff+1
    v2f amat[4];
#pragma unroll
    for (int kk = 0; kk < 4; ++kk) {
        amat[kk].x = ((4 * kk + koff)     <= n) ? 1.0f : 0.0f;
        amat[kk].y = ((4 * kk + koff + 1) <= n) ? 1.0f : 0.0f;
    }

    for (int mc = 0; mc < 4; ++mc) {           // 4 * 256 = 1024 >= 788 output steps
        const int t0 = mc * 256;

        // prefetch start of next macro-chunk's input window (streaming hint)
        {
            const int nidx = (int)((float)(t0 + 256) * RATE_F);
            if (nidx < T_IN) __builtin_prefetch(&spec2[nidx], 0, 0);
        }

        float mag[8];
#pragma unroll
        for (int i = 0; i < 8; ++i) {
            const int   t     = t0 + i * 32 + lane;
            const float ts    = (float)t * RATE_F;
            const int   idx   = (int)ts;
            const float alpha = ts - (float)idx;

            float2 s0 = make_float2(0.f, 0.f);
            float2 s1 = make_float2(0.f, 0.f);
            if (idx     < T_IN) s0 = spec2[idx];
            if (idx + 1 < T_IN) s1 = spec2[idx + 1];

            const float ang0 = atan2f(s0.y, s0.x);
            const float ang1 = atan2f(s1.y, s1.x);
            const float nr0  = sqrtf(s0.x * s0.x + s0.y * s0.y);
            const float nr1  = sqrtf(s1.x * s1.x + s1.y * s1.y);
            mag[i] = alpha * nr1 + (1.0f - alpha) * nr0;

            float d = ang1 - ang0 - padv;
            d = d - TWO_PI * rintf(d * INV_TWO_PI);   // RNE wrap == jnp.round
            d = d + padv;

            // p[t] = concat(phase_0, delta[:-1])[t]  -> shift delta up by one
            float p = __shfl_up(d, 1, 32);
            if (lane == 0) p = prev_delta;
            prev_delta = __shfl(d, 31, 32);

            pbuf[i * 32 + lane] = p;
        }
        asm volatile("s_wait_dscnt 0" ::: "memory");

        // Build B fragments: X[k][col] = p[col*16 + k]; 16 columns = 16 segments.
        // B layout (4x16): v.x -> K = 4kk+koff (N = lane&15), v.y -> K = 4kk+koff+1
        v2f bmat[4];
#pragma unroll
        for (int kk = 0; kk < 4; ++kk) {
            bmat[kk].x = pbuf[n * 16 + 4 * kk + koff];
            bmat[kk].y = pbuf[n * 16 + 4 * kk + koff + 1];
        }

        // Y = L(16x16 lower-tri ones) @ X : 16 parallel inclusive scans of 16
        v8f c = {};
#pragma unroll
        for (int kk = 0; kk < 4; ++kk) {
            c = __builtin_amdgcn_wmma_f32_16x16x4_f32(
                    false, amat[kk], false, bmat[kk],
                    (short)0, c, false, false);
        }

        // Column totals: Y[15][col] lives in VGPR 7 of lanes 16..31
        if (hi) tbuf[n] = c[7];
        asm volatile("s_wait_dscnt 0" ::: "memory");

        float off = 0.0f, sum16 = 0.0f;
#pragma unroll
        for (int j = 0; j < 16; ++j) {
            const float tv = tbuf[j];
            sum16 += tv;
            if (j < n) off += tv;    // exclusive prefix over segment totals
        }
        const float base = off + carry;
        carry += sum16;

        // Scatter phase_acc back into time order (C/D: lane holds col n, rows mrow_base+v)
#pragma unroll
        for (int v = 0; v < 8; ++v) {
            pbuf[n * 16 + mrow_base + v] = c[v] + base;
        }
        asm volatile("s_wait_dscnt 0" ::: "memory");

#pragma unroll
        for (int i = 0; i < 8; ++i) {
            const int t = t0 + i * 32 + lane;
            if (t < T_OUT) {
                const float pa = pbuf[i * 32 + lane];
                float sn, cs;
                __sincosf(pa, &sn, &cs);
                out2[t] = make_float2(mag[i] * cs, mag[i] * sn);
            }
        }
    }
}

extern "C" void kernel_launch(void* const* d_in, const int* in_sizes, int n_in,
                              void* d_out, int out_size, void* d_ws, size_t ws_size,
                              hipStream_t stream) {
    const float* spec = (const float*)d_in[0];   // (32,1,1025,1024,2) f32
    const float* padv = (const float*)d_in[1];   // (1025,1) f32
    float* out = (float*)d_out;                  // (32,1,1025,788,2) f32

    const int blocks = ROWS / WAVES_PER_BLOCK;   // 4100, exact
    phase_vocoder_kernel<<<blocks, 256, 0, stream>>>(spec, padv, out);
}